// DeChunkLayer_63917703299657
// MI455X (gfx1250) — compile-verified
//
#include <hip/hip_runtime.h>
#include <hip/hip_bf16.h>

typedef __attribute__((ext_vector_type(2))) float v2f;
typedef __attribute__((ext_vector_type(8))) float v8f;

// ---------------------------------------------------------------------------
// Kernel 1: per-batch inclusive cumsum of chunk_lengths  (C = 2048, 256 thr)
// ---------------------------------------------------------------------------
__global__ __launch_bounds__(256)
void cumsum_kernel(const int* __restrict__ lens, int* __restrict__ cum, int C) {
    __shared__ int sdata[256];
    const int b = blockIdx.x;
    const int tid = threadIdx.x;
    constexpr int PER = 8;                 // C / 256 == 2048 / 256
    const int* lb = lens + b * C;
    int*       cb = cum  + b * C;

    int local[PER];
    int s = 0;
#pragma unroll
    for (int j = 0; j < PER; ++j) { s += lb[tid * PER + j]; local[j] = s; }
    sdata[tid] = s;
    __syncthreads();

    // Hillis-Steele inclusive scan over 256 thread totals
    for (int off = 1; off < 256; off <<= 1) {
        int v = (tid >= off) ? sdata[tid - off] : 0;
        __syncthreads();
        sdata[tid] += v;
        __syncthreads();
    }
    const int excl = sdata[tid] - s;
#pragma unroll
    for (int j = 0; j < PER; ++j) cb[tid * PER + j] = excl + local[j];
}

// ---------------------------------------------------------------------------
// Kernel 2: proj = X * W^T + bias  via V_WMMA_F32_16X16X4_F32
//   M = 16384 (B*C), N = K = 512.
//   Block = 256 threads = 8 waves. Block covers 32 rows x full N=512.
//   wave w: rows [block*32 + (w/4)*16, +16), cols [(w%4)*128, +128)
//   -> 8 WMMA accumulators per wave, A fragment reused across all 8.
// ---------------------------------------------------------------------------
__global__ __launch_bounds__(256)
void gemm_wmma_kernel(const float* __restrict__ X, const float* __restrict__ W,
                      const float* __restrict__ bias, float* __restrict__ proj) {
    constexpr int K = 512;
    constexpr int N = 512;

    const int lane   = threadIdx.x & 31;
    const int wave   = threadIdx.x >> 5;
    const int m_sub  = wave >> 2;          // 0..1
    const int n_quad = wave & 3;           // 0..3

    // fp32 WMMA 16x16x4 fragment addressing (ISA 7.12.2):
    //   A: lanes 0-15 -> M=lane,     V0=K0,V1=K1 ; lanes 16-31 -> M=lane-16, V0=K2,V1=K3
    //   B: symmetric for 4x16 (per-lane column, 2 consecutive K values)
    const int mlane  = lane & 15;
    const int khalf  = (lane >> 4) * 2;

    const int row    = blockIdx.x * 32 + m_sub * 16 + mlane;     // X row
    const float* xrow = X + (size_t)row * K;

    v8f acc[8] = {};   // zero accumulators

#pragma unroll 2
    for (int k0 = 0; k0 < K; k0 += 4) {
        const int kk = k0 + khalf;
        v2f a = *(const v2f*)(xrow + kk);
#pragma unroll
        for (int t = 0; t < 8; ++t) {
            const int ncol = n_quad * 128 + t * 16 + mlane;      // W row (== output col)
            v2f bf = *(const v2f*)(W + (size_t)ncol * K + kk);
            acc[t] = __builtin_amdgcn_wmma_f32_16x16x4_f32(
                false, a, false, bf, (short)0, acc[t], false, false);
        }
    }

    // C/D layout: VGPR v, lanes 0-15 -> M=v, lanes 16-31 -> M=v+8 ; N=lane&15
    const int mrow = blockIdx.x * 32 + m_sub * 16 + 8 * (lane >> 4);
#pragma unroll
    for (int t = 0; t < 8; ++t) {
        const int n = n_quad * 128 + t * 16 + mlane;
        const float bv = bias[n];
#pragma unroll
        for (int v = 0; v < 8; ++v) {
            proj[(size_t)(mrow + v) * N + n] = acc[t][v] + bv;
        }
    }
}

// ---------------------------------------------------------------------------
// Kernel 3: per-row mean / rstd over D=512 (one wave per row, 8 rows/block)
// ---------------------------------------------------------------------------
__global__ __launch_bounds__(256)
void row_stats_kernel(const float* __restrict__ proj,
                      float* __restrict__ mean, float* __restrict__ rstd) {
    const int lane = threadIdx.x & 31;
    const int wave = threadIdx.x >> 5;
    const int r = blockIdx.x * 8 + wave;
    const float* p = proj + (size_t)r * 512;

    float s = 0.f, s2 = 0.f;
#pragma unroll
    for (int j = 0; j < 4; ++j) {
        float4 x = *(const float4*)(p + lane * 4 + j * 128);
        s  += x.x + x.y + x.z + x.w;
        s2 += x.x * x.x + x.y * x.y + x.z * x.z + x.w * x.w;
    }
#pragma unroll
    for (int off = 16; off > 0; off >>= 1) {
        s  += __shfl_xor(s,  off, 32);
        s2 += __shfl_xor(s2, off, 32);
    }
    if (lane == 0) {
        const float m   = s * (1.f / 512.f);
        const float var = s2 * (1.f / 512.f) - m * m;
        mean[r] = m;
        rstd[r] = rsqrtf(var + 1e-5f);
    }
}

// ---------------------------------------------------------------------------
// Kernel 4: gather chunk rows by searchsorted(cum, t, 'right') + apply LN
//   One wave per output row; lane 0 binary-searches, broadcasts via shfl.
// ---------------------------------------------------------------------------
__global__ __launch_bounds__(256)
void gather_ln_kernel(const float* __restrict__ proj,
                      const float* __restrict__ mean, const float* __restrict__ rstd,
                      const int* __restrict__ cum,
                      const float* __restrict__ gamma, const float* __restrict__ beta,
                      float* __restrict__ out, int C, int L) {
    const int lane = threadIdx.x & 31;
    const int wave = threadIdx.x >> 5;
    const int g = blockIdx.x * 8 + wave;       // global output row in [0, B*L)
    const int b = g / L;
    const int t = g - b * L;
    const int* cb = cum + b * C;

    int idx = 0;
    if (lane == 0) {
        int lo = 0, hi = C;
        while (lo < hi) {                      // first i with cum[i] > t
            int mid = (lo + hi) >> 1;
            if (cb[mid] <= t) lo = mid + 1; else hi = mid;
        }
        idx = lo;
    }
    idx = __shfl(idx, 0, 32);

    const int  total = cb[C - 1];
    const bool valid = t < total;
    const int  ci  = idx < (C - 1) ? idx : (C - 1);
    const int  srow = b * C + ci;

    const float m = mean[srow];
    const float r = rstd[srow];
    const float* p = proj + (size_t)srow * 512;
    float* o = out + (size_t)g * 512;

#pragma unroll
    for (int j = 0; j < 4; ++j) {
        const int d = lane * 4 + j * 128;
        float4 gm = *(const float4*)(gamma + d);
        float4 bt = *(const float4*)(beta + d);
        float4 y;
        if (valid) {
            float4 x = *(const float4*)(p + d);
            y.x = (x.x - m) * r * gm.x + bt.x;
            y.y = (x.y - m) * r * gm.y + bt.y;
            y.z = (x.z - m) * r * gm.z + bt.z;
            y.w = (x.w - m) * r * gm.w + bt.w;
        } else {
            y = bt;                            // LN(zero row) = beta
        }
        *(float4*)(o + d) = y;
    }
}

// ---------------------------------------------------------------------------
extern "C" void kernel_launch(void* const* d_in, const int* in_sizes, int n_in,
                              void* d_out, int out_size, void* d_ws, size_t ws_size,
                              hipStream_t stream) {
    (void)n_in; (void)ws_size;

    const float* X     = (const float*)d_in[0];   // [B, C, D]
    const float* W     = (const float*)d_in[1];   // [D, D]
    const float* bias  = (const float*)d_in[2];   // [D]
    const float* gamma = (const float*)d_in[3];   // [D]
    const float* beta  = (const float*)d_in[4];   // [D]
    const int*   lens  = (const int*)d_in[5];     // [B, C]
    // d_in[6] = seq_len lives on device; derived below instead.

    const int Dm = in_sizes[2];                   // 512
    const int BC = in_sizes[5];                   // B*C = 16384
    const int B  = 8;
    const int C  = BC / B;                        // 2048
    const int L  = out_size / (B * Dm);           // 8192

    char*  ws   = (char*)d_ws;
    int*   cum  = (int*)ws;                       // BC ints      (64 KB)
    float* mean = (float*)(ws + (64 << 10));      // BC floats    (64 KB)
    float* rstd = (float*)(ws + (128 << 10));     // BC floats    (64 KB)
    float* proj = (float*)(ws + (192 << 10));     // BC*Dm floats (33.5 MB)
    float* out  = (float*)d_out;

    cumsum_kernel  <<<B,            256, 0, stream>>>(lens, cum, C);
    gemm_wmma_kernel<<<BC / 32,     256, 0, stream>>>(X, W, bias, proj);
    row_stats_kernel<<<BC / 8,      256, 0, stream>>>(proj, mean, rstd);
    gather_ln_kernel<<<(B * L) / 8, 256, 0, stream>>>(proj, mean, rstd, cum,
                                                      gamma, beta, out, C, L);
}